// RetinaNetFocalLoss_59468117180785
// MI455X (gfx1250) — compile-verified
//
#include <hip/hip_runtime.h>
#include <cmath>

#define NB 16
#define NA 49104
#define NC 80
#define NT 64

typedef __attribute__((ext_vector_type(2))) float v2f;
typedef __attribute__((ext_vector_type(8))) float v8f;

// aspects = [s*sqrt(r), s*sqrt(1/r)] for r in {0.5,1,2}, s in {1, 2^-1/3, 2^-2/3}
__constant__ float c_aspect_h[9] = {
    0.7071067811865476f, 0.5612310241546865f, 0.4454493590701698f,
    1.0f,                0.7937005259840997f, 0.6299605249474366f,
    1.4142135623730951f, 1.1224620483093730f, 0.8908987181403393f};
__constant__ float c_aspect_w[9] = {
    1.4142135623730951f, 1.1224620483093730f, 0.8908987181403393f,
    1.0f,                0.7937005259840997f, 0.6299605249474366f,
    0.7071067811865476f, 0.5612310241546865f, 0.4454493590701698f};

// ---------------------------------------------------------------------------
// Wave32 sum via V_WMMA_F32_16X16X4_F32.
// A = ones(16x4). B holds each lane's value exactly once (B vgpr0 = s,
// vgpr1 = 0), so D[m][n] = colsum_B(n) and sum_n colsum(n) = sum of all lane
// values.  By the documented C/D layout, D's vgpr0 = row 0 over cols 0..15
// (lanes 0-15) and row 8 over cols 0..15 (lanes 16-31): each 16-lane half of
// c[0] contains every column-sum exactly once.  A 4-step xor butterfly with
// offsets 8,4,2,1 (never crossing the half) therefore leaves the full total
// in every lane.  Must be reached by all lanes (EXEC all ones).
// ---------------------------------------------------------------------------
__device__ __forceinline__ float wave_sum_wmma(float s) {
  v2f a; a[0] = 1.0f; a[1] = 1.0f;
  v2f bm; bm[0] = s;  bm[1] = 0.0f;
  v8f c = {};
  c = __builtin_amdgcn_wmma_f32_16x16x4_f32(false, a, false, bm,
                                            (short)0, c, false, false);
  float v = c[0];
#pragma unroll
  for (int off = 8; off >= 1; off >>= 1) v += __shfl_xor(v, off, 32);
  return v;
}

__device__ __forceinline__ void atomic_add_f32(float* p, float v) {
  unsafeAtomicAdd(p, v);   // global_atomic_add_f32 (no CAS loop)
}

__device__ __forceinline__ float smooth_l1(float d) {
  d = fabsf(d);
  return (d < 1.0f) ? 0.5f * d * d : d - 0.5f;
}

__device__ __forceinline__ void decode_anchor(int i, float& cy, float& cx,
                                              float& ah, float& aw) {
  int start, hw;
  if (i < 36864)      { start = 0;     hw = 64; }
  else if (i < 46080) { start = 36864; hw = 32; }
  else if (i < 48384) { start = 46080; hw = 16; }
  else if (i < 48960) { start = 48384; hw = 8;  }
  else                { start = 48960; hw = 4;  }
  int il = i - start;
  int n  = il / 9;
  int a  = il - n * 9;
  int iy = n / hw;
  int ix = n - iy * hw;
  float inv = 1.0f / (float)hw;                    // power of two: exact
  cy = (float)(2 * iy + 1) * inv - 1.0f;           // exact, matches numpy grid
  cx = (float)(2 * ix + 1) * inv - 1.0f;
  ah = c_aspect_h[a] * (2.0f * inv);
  aw = c_aspect_w[a] * (2.0f * inv);
}

__global__ void init_kernel(float* acc) {
  int t = threadIdx.x;
  if (t < 48) acc[t] = 0.0f;  // accBB[16] | accN[16] | accCL[16]
}

// ---------------------------------------------------------------------------
// Kernel 1: matching + bbox smooth-L1. grid = (ceil(NA/256), NB)
// ---------------------------------------------------------------------------
__global__ void match_kernel(const float* __restrict__ bbox_preds,
                             const float* __restrict__ bbox_tgts,
                             const int*   __restrict__ clas_tgts,
                             int*   __restrict__ matches,
                             float* __restrict__ acc) {
  __shared__ float s_cy[NT], s_cx[NT], s_h[NT], s_w[NT], s_area[NT];
  __shared__ int   s_cls[NT];
  __shared__ int   s_any;
  __shared__ float s_red0[8], s_red1[8];

  const int b   = blockIdx.y;
  const int tid = threadIdx.x;

  if (tid == 0) s_any = 0;
  __syncthreads();
  if (tid < NT) {
    const float4 tb = ((const float4*)bbox_tgts)[(size_t)b * NT + tid]; // tlbr
    float cy = (tb.x + tb.z) * 0.5f;   // tlbr -> cthw (matches reference path)
    float cx = (tb.y + tb.w) * 0.5f;
    float h  = tb.z - tb.x;
    float w  = tb.w - tb.y;
    s_cy[tid] = cy; s_cx[tid] = cx; s_h[tid] = h; s_w[tid] = w;
    s_area[tid] = h * w;
    int cls = clas_tgts[b * NT + tid];
    s_cls[tid] = cls;
    if (cls != 0) atomicOr(&s_any, 1);
  }
  __syncthreads();

  const int i = blockIdx.x * blockDim.x + tid;
  float bb = 0.0f, nm = 0.0f;
  if (i < NA) {
    float cy, cx, ah, aw;
    decode_anchor(i, cy, cx, ah, aw);
    const float a_tly = cy - ah * 0.5f, a_tlx = cx - aw * 0.5f;
    const float a_bry = cy + ah * 0.5f, a_brx = cx + aw * 0.5f;
    const float a_area = ah * aw;

    float best = -2.0f;
    int bidx = 0;
#pragma unroll 8
    for (int t = 0; t < NT; ++t) {
      float th = s_h[t], tw = s_w[t];
      float ttly = s_cy[t] - th * 0.5f, ttlx = s_cx[t] - tw * 0.5f;
      float tbry = s_cy[t] + th * 0.5f, tbrx = s_cx[t] + tw * 0.5f;
      float ih = fminf(a_bry, tbry) - fmaxf(a_tly, ttly);
      float iw = fminf(a_brx, tbrx) - fmaxf(a_tlx, ttlx);
      ih = fmaxf(ih, 0.0f); iw = fmaxf(iw, 0.0f);
      float inter = ih * iw;
      float iou = inter / (a_area + s_area[t] - inter + 1e-8f);
      iou = (s_cls[t] != 0) ? iou : -1.0f;
      if (iou > best) { best = iou; bidx = t; }   // first-max (strict >)
    }

    int code = -2;                                 // ignore
    bool matched = false;
    if (s_any) {
      if (best > 0.5f)      { code = s_cls[bidx] - 1; matched = true; }
      else if (best < 0.4f) { code = -1; }         // background
    }
    matches[(size_t)b * NA + i] = code;

    if (matched) {
      float a0 = ((s_cy[bidx] - cy) / ah) / 0.1f;
      float a1 = ((s_cx[bidx] - cx) / aw) / 0.1f;
      float a2 = logf(s_h[bidx] / ah + 1e-8f) / 0.2f;
      float a3 = logf(s_w[bidx] / aw + 1e-8f) / 0.2f;
      const float4 p = ((const float4*)bbox_preds)[(size_t)b * NA + i];
      bb = smooth_l1(p.x - a0) + smooth_l1(p.y - a1) +
           smooth_l1(p.z - a2) + smooth_l1(p.w - a3);
      nm = 1.0f;
    }
  }

  float wbb = wave_sum_wmma(bb);   // full EXEC: every thread reaches here
  float wnm = wave_sum_wmma(nm);
  const int wid = tid >> 5, lane = tid & 31;
  if (lane == 0) { s_red0[wid] = wbb; s_red1[wid] = wnm; }
  __syncthreads();
  if (tid == 0) {
    float tb = 0.0f, tn = 0.0f;
#pragma unroll
    for (int k = 0; k < 8; ++k) { tb += s_red0[k]; tn += s_red1[k]; }
    atomic_add_f32(&acc[b], tb);        // accBB
    atomic_add_f32(&acc[16 + b], tn);   // accN
  }
}

// ---------------------------------------------------------------------------
// Kernel 2: focal BCE, one thread per (b, anchor, class). grid = (NA*NC/256, NB)
// ---------------------------------------------------------------------------
__global__ void focal_kernel(const float* __restrict__ clas_preds,
                             const int*   __restrict__ matches,
                             float* __restrict__ acc) {
  __shared__ float s_red[8];
  const int b   = blockIdx.y;
  const int tid = threadIdx.x;
  const int e   = blockIdx.x * 256 + tid;      // < NA*NC exactly
  const int i   = e / NC;
  const int c   = e - i * NC;

  float contrib = 0.0f;
  const int m = matches[(size_t)b * NA + i];
  if (m >= -1) {                                // not ignored
    const float x  = clas_preds[(size_t)b * NA * NC + e];
    const float ax = fabsf(x);
    const float t  = expf(-ax);
    const float l1p = log1pf(t);                 // log(1+e^{-|x|})
    const float sig_ax = 1.0f / (1.0f + t);      // sigmoid(|x|)
    const float ps = (x >= 0.0f) ? sig_ax : (1.0f - sig_ax);
    if (m == c) {                                // matched target class
      const float omp = 1.0f - ps;
      const float sp_neg = fmaxf(-x, 0.0f) + l1p;   // softplus(-x) = -logsig(x)
      contrib = 0.75f * omp * omp * sp_neg;
    } else {                                     // background / other class
      const float sp_pos = fmaxf(x, 0.0f) + l1p;    // softplus(x) = -logsig(-x)
      contrib = 0.25f * ps * ps * sp_pos;
    }
  }

  float ws = wave_sum_wmma(contrib);
  const int wid = tid >> 5, lane = tid & 31;
  if (lane == 0) s_red[wid] = ws;
  __syncthreads();
  if (tid == 0) {
    float tot = 0.0f;
#pragma unroll
    for (int k = 0; k < 8; ++k) tot += s_red[k];
    atomic_add_f32(&acc[32 + b], tot);           // accCL
  }
}

// ---------------------------------------------------------------------------
// Kernel 3: finalize — per-batch losses, mean over batch.
// ---------------------------------------------------------------------------
__global__ void finalize_kernel(const float* __restrict__ acc,
                                float* __restrict__ out) {
  const int l = threadIdx.x;
  float v = 0.0f;
  if (l < NB) {
    const float n  = acc[16 + l];
    const float bb = acc[l]      / fmaxf(4.0f * n, 1.0f);  // 0 when n==0
    const float cl = acc[32 + l] / fmaxf(n, 1.0f);
    v = bb + cl;
  }
#pragma unroll
  for (int off = 16; off >= 1; off >>= 1) v += __shfl_xor(v, off, 32);
  if (l == 0) out[0] = v / (float)NB;
}

extern "C" void kernel_launch(void* const* d_in, const int* in_sizes, int n_in,
                              void* d_out, int out_size, void* d_ws, size_t ws_size,
                              hipStream_t stream) {
  const float* clas_preds = (const float*)d_in[0];   // [16, 49104, 80]
  const float* bbox_preds = (const float*)d_in[1];   // [16, 49104, 4]
  const float* bbox_tgts  = (const float*)d_in[2];   // [16, 64, 4] tlbr
  const int*   clas_tgts  = (const int*)d_in[3];     // [16, 64]

  float* acc     = (float*)d_ws;                     // 48 floats: BB|N|CL
  int*   matches = (int*)((char*)d_ws + 256);        // 16*49104 ints

  init_kernel<<<1, 64, 0, stream>>>(acc);

  dim3 g1((NA + 255) / 256, NB);
  match_kernel<<<g1, 256, 0, stream>>>(bbox_preds, bbox_tgts, clas_tgts,
                                       matches, acc);

  dim3 g2((NA * NC) / 256, NB);                      // 15345 x 16, exact
  focal_kernel<<<g2, 256, 0, stream>>>(clas_preds, matches, acc);

  finalize_kernel<<<1, 32, 0, stream>>>(acc, (float*)d_out);
}